// SigmoidGatedCRFLoss_858993459295
// MI455X (gfx1250) — compile-verified
//
#include <hip/hip_runtime.h>
#include <cstdint>

// ---------------- problem constants (reference: N=2, H=W=256) ----------------
constexpr int IMG_H  = 256;
constexpr int IMG_W  = 256;
constexpr int NBATCH = 2;
constexpr int RAD    = 5;
constexpr int DIA    = 11;          // 2*RAD+1
constexpr int TX     = 32;          // tile width  (pixels per block)
constexpr int TY     = 8;           // tile height
constexpr int HALO_W = TX + 2 * RAD;   // 42
constexpr int HALO_H = TY + 2 * RAD;   // 18
constexpr int LDSW   = 44;             // padded LDS row stride (floats / float4s)
constexpr int NBLOCKS = (IMG_W / TX) * (IMG_H / TY) * NBATCH; // 512

typedef float v2f __attribute__((ext_vector_type(2)));
typedef float v8f __attribute__((ext_vector_type(8)));

// binarize per reference: NaN -> 0, v < 1 -> 0, else keep value.
// (NaN >= 1.0f) is false, so this handles NaN without an explicit isnan.
__device__ __forceinline__ float binarize(float v) {
    return (v >= 1.0f) ? v : 0.0f;
}

// Async DMA: global -> LDS, 4 bytes, per-lane addresses. Tracked by ASYNCcnt.
// LDS byte address = low 32 bits of the flat pointer (LDS aperture layout).
__device__ __forceinline__ void async_copy_b32(void* lds_dst, const float* gsrc) {
    unsigned laddr = (unsigned)(unsigned long long)lds_dst;
    unsigned long long gaddr = (unsigned long long)gsrc;
    asm volatile("global_load_async_to_lds_b32 %0, %1, off"
                 :: "v"(laddr), "v"(gaddr)
                 : "memory");
}

__device__ __forceinline__ void wait_asynccnt0() {
    asm volatile("s_wait_asynccnt 0" ::: "memory");
}

// ---------------------------------------------------------------------------
// Kernel 1: per-pixel 11x11 gated-CRF stencil, per-block deterministic
// reduction. Writes part[bid] = sum(contrib), part[NBLOCKS+bid] = sum(md).
// ---------------------------------------------------------------------------
__global__ __launch_bounds__(TX * TY) void crf_main_kernel(
    const float* __restrict__ x,     // [N,3,H,W]
    const float* __restrict__ y,     // [N,H,W]
    const float* __restrict__ msrc,  // [N,1,H,W]
    const float* __restrict__ mdst,  // [N,1,H,W]
    float* __restrict__ part)        // [2*NBLOCKS]
{
    __shared__ float4 s_feat[HALO_H * LDSW];   // {x0, x1, x2, sigmoid(y)}
    __shared__ float  s_ms  [HALO_H * LDSW];   // binarized mask_src (0 at OOB)
    __shared__ float  s_spat[DIA * DIA];       // exp(-(dx^2+dy^2)/72)
    __shared__ float  red0[TX * TY];
    __shared__ float  red1[TX * TY];

    const int n   = blockIdx.z;
    const int by  = blockIdx.y * TY;
    const int bx  = blockIdx.x * TX;
    const int tx  = threadIdx.x;
    const int ty  = threadIdx.y;
    const int tid = ty * TX + tx;

    // spatial affinity table (121 entries)
    if (tid < DIA * DIA) {
        int dy = tid / DIA - RAD;
        int dx = tid % DIA - RAD;
        s_spat[tid] = __expf(-(float)(dx * dx + dy * dy) * (1.0f / 72.0f));
    }

    const float* xn  = x    + (size_t)n * 3 * IMG_H * IMG_W;
    const float* yn  = y    + (size_t)n * IMG_H * IMG_W;
    const float* msn = msrc + (size_t)n * IMG_H * IMG_W;
    const float* mdn = mdst + (size_t)n * IMG_H * IMG_W;

    // ---- stage halo tile: x via async DMA, sigmoid(y)/mask via VALU + DS ----
    for (int i = tid; i < HALO_H * HALO_W; i += TX * TY) {
        const int r  = i / HALO_W;
        const int c  = i % HALO_W;
        const int gh = by + r - RAD;
        const int gw = bx + c - RAD;
        const int li = r * LDSW + c;
        if (gh >= 0 && gh < IMG_H && gw >= 0 && gw < IMG_W) {
            const int gidx = gh * IMG_W + gw;
            float* slot = (float*)&s_feat[li];
            async_copy_b32(slot + 0, xn + 0 * IMG_H * IMG_W + gidx);
            async_copy_b32(slot + 1, xn + 1 * IMG_H * IMG_W + gidx);
            async_copy_b32(slot + 2, xn + 2 * IMG_H * IMG_W + gidx);
            const float yv = yn[gidx];
            slot[3]  = 1.0f / (1.0f + __expf(-yv));   // sigmoid
            s_ms[li] = binarize(msn[gidx]);
        } else {
            s_feat[li] = make_float4(0.0f, 0.0f, 0.0f, 0.0f);
            s_ms[li]   = 0.0f;   // zero-padded unfold(mask_src) kills OOB taps
        }
    }
    wait_asynccnt0();
    __syncthreads();

    // ---- 11x11 stencil for this thread's pixel (always in-bounds) ----------
    const int    lc = (ty + RAD) * LDSW + (tx + RAD);
    const float4 fp = s_feat[lc];
    const float  sp = fp.w;
    const float  mdp = binarize(mdn[(by + ty) * IMG_W + (bx + tx)]);

    float acc = 0.0f;
    for (int dy = 0; dy < DIA; ++dy) {
#pragma unroll
        for (int dx = 0; dx < DIA; ++dx) {
            if (dy == RAD && dx == RAD) continue;          // center zeroed
            const int    li  = (ty + dy) * LDSW + (tx + dx);
            const float  msq = s_ms[li];
            const float4 fq  = s_feat[li];
            const float d0 = fq.x - fp.x;
            const float d1 = fq.y - fp.y;
            const float d2 = fq.z - fp.z;
            const float dd = d0 * d0 + d1 * d1 + d2 * d2;
            // 0.9*exp(-0.5*(S+100*dd)) + 0.1*exp(-0.5*S)
            const float wgt = s_spat[dy * DIA + dx] *
                              (0.9f * __expf(-50.0f * dd) + 0.1f);
            const float e = sp + fq.w - 2.0f * sp * fq.w;  // 1-(1-sq)(1-sp)-sq*sp
            acc += wgt * msq * e;
        }
    }
    const float contrib = mdp * acc;

    // ---- deterministic block tree reduction (no float atomics) -------------
    red0[tid] = contrib;
    red1[tid] = mdp;
    __syncthreads();
    for (int s = (TX * TY) / 2; s > 0; s >>= 1) {
        if (tid < s) {
            red0[tid] += red0[tid + s];
            red1[tid] += red1[tid + s];
        }
        __syncthreads();
    }
    if (tid == 0) {
        const int bid = blockIdx.x +
                        gridDim.x * (blockIdx.y + gridDim.y * blockIdx.z);
        part[bid]           = red0[0];
        part[NBLOCKS + bid] = red1[0];
    }
}

// ---------------------------------------------------------------------------
// Kernel 2: single-wave WMMA reduction of the 2x512 partials.
// A = ones(16x4)  =>  D = A*B + C accumulates column sums of B; summing row 0
// of the final accumulator yields the grand total. Deterministic order.
// ---------------------------------------------------------------------------
__global__ __launch_bounds__(32) void crf_reduce_kernel(
    const float* __restrict__ part, float* __restrict__ out)
{
    const int lane = threadIdx.x;
    __shared__ float tmp[16];
    __shared__ float totals[2];

    for (int a = 0; a < 2; ++a) {
        const float* base = part + a * NBLOCKS;
        v8f c = {};                       // C/D accumulator (16x16 f32)
        v2f A;  A.x = 1.0f; A.y = 1.0f;   // 16x4 all-ones
        for (int it = 0; it < NBLOCKS; it += 64) {
            v2f B;                        // 4x16 chunk of 64 partials
            B.x = base[it + lane];
            B.y = base[it + 32 + lane];
            c = __builtin_amdgcn_wmma_f32_16x16x4_f32(
                    /*neg_a=*/false, A, /*neg_b=*/false, B,
                    /*c_mod=*/(short)0, c,
                    /*reuse_a=*/false, /*reuse_b=*/false);
        }
        // Row M=0 lives in element 0 of lanes 0..15 (N = lane).
        if (lane < 16) tmp[lane] = c[0];
        __syncthreads();
        if (lane == 0) {
            float s = 0.0f;
            for (int i = 0; i < 16; ++i) s += tmp[i];
            totals[a] = s;
        }
        __syncthreads();
    }
    if (lane == 0) {
        const float denom = fmaxf(totals[1], 1.0f);
        out[0] = totals[0] / denom;
    }
}

// ---------------------------------------------------------------------------
extern "C" void kernel_launch(void* const* d_in, const int* in_sizes, int n_in,
                              void* d_out, int out_size, void* d_ws, size_t ws_size,
                              hipStream_t stream) {
    (void)in_sizes; (void)n_in; (void)out_size; (void)ws_size;
    const float* x    = (const float*)d_in[0];
    const float* y    = (const float*)d_in[1];
    const float* msrc = (const float*)d_in[2];
    const float* mdst = (const float*)d_in[3];
    float* part = (float*)d_ws;          // 2*NBLOCKS floats = 4 KB scratch

    dim3 grid(IMG_W / TX, IMG_H / TY, NBATCH);   // (8, 32, 2) = 512 blocks
    dim3 block(TX, TY, 1);                       // 256 threads = 8 waves
    crf_main_kernel<<<grid, block, 0, stream>>>(x, y, msrc, mdst, part);
    crf_reduce_kernel<<<1, 32, 0, stream>>>(part, (float*)d_out);
}